// GNNModel_53120155517092
// MI455X (gfx1250) — compile-verified
//
#include <hip/hip_runtime.h>

// GCN 2-layer forward for MI455X (gfx1250, wave32).
// Dense transforms use native fp32 WMMA (v_wmma_f32_16x16x4_f32).
// Edge aggregation uses L2-resident float atomics (tables fit in 192MB L2).

typedef float v2f __attribute__((ext_vector_type(2)));
typedef float v8f __attribute__((ext_vector_type(8)));

#define IN_CH 128
#define HID   16
#define OUT_CH 32

// ---------- degree / normalization ----------

__global__ __launch_bounds__(256) void k_init_deg(float* __restrict__ deg, int N) {
  int i = blockIdx.x * blockDim.x + threadIdx.x;
  if (i < N) deg[i] = 1.0f;  // self-loop contribution
}

__global__ __launch_bounds__(256) void k_count_deg(const long long* __restrict__ ei,
                                                   float* __restrict__ deg, long long E) {
  long long e = (long long)blockIdx.x * blockDim.x + threadIdx.x;
  if (e < E) {
    long long d = ei[E + e];  // dst row
    atomicAdd(&deg[d], 1.0f);
  }
}

__global__ __launch_bounds__(256) void k_rsqrt_deg(float* __restrict__ deg, int N) {
  int i = blockIdx.x * blockDim.x + threadIdx.x;
  if (i < N) {
    float v = deg[i];
    deg[i] = (v > 0.0f) ? rsqrtf(v) : 0.0f;  // deg >= 1 with self-loops
  }
}

// ---------- layer 1 GEMM: t1[N,16] = x[N,128] @ W1[128,16] (fp32 WMMA) ----------
// One wave per 16-row tile. A 16x4 f32 layout: lanes 0-15 hold K=k0,k0+1 in
// a[0],a[1]; lanes 16-31 hold K=k0+2,k0+3. B 4x16 mirrored. C/D: vgpr r ->
// (M=r, N=lane) for lanes 0-15 and (M=r+8, N=lane-16) for lanes 16-31.

__global__ __launch_bounds__(256) void k_gemm1_wmma(const float* __restrict__ x,
                                                    const float* __restrict__ W1,
                                                    float* __restrict__ t1, int nTiles) {
  int wave = (int)((blockIdx.x * blockDim.x + threadIdx.x) >> 5);
  if (wave >= nTiles) return;                 // uniform per wave: EXEC stays all-1s
  int lane = threadIdx.x & 31;
  int m  = lane & 15;
  int hi = lane >> 4;

  v8f c = {};
  const float* __restrict__ arow = x + (size_t)(wave * 16 + m) * IN_CH;
  #pragma unroll 4
  for (int kk = 0; kk < IN_CH / 4; ++kk) {
    int k0 = kk * 4 + hi * 2;
    v2f a, b;
    a[0] = arow[k0];
    a[1] = arow[k0 + 1];
    b[0] = W1[(size_t)k0 * HID + m];
    b[1] = W1[(size_t)(k0 + 1) * HID + m];
    c = __builtin_amdgcn_wmma_f32_16x16x4_f32(false, a, false, b, (short)0, c, false, false);
  }
  float* __restrict__ orow = t1 + (size_t)(wave * 16 + hi * 8) * HID + m;
  #pragma unroll
  for (int r = 0; r < 8; ++r) orow[(size_t)r * HID] = c[r];
}

// ---------- layer 1 aggregation ----------
// agg1[i,c] = b1[c] + t1[i,c]*dis[i]^2   (bias + self-loop, also initializes buffer)
__global__ __launch_bounds__(256) void k_self1(const float* __restrict__ t1,
                                               const float* __restrict__ dis,
                                               const float* __restrict__ b1,
                                               float* __restrict__ agg1, long long total) {
  long long t = (long long)blockIdx.x * blockDim.x + threadIdx.x;
  if (t < total) {
    long long i = t >> 4;
    int ch = (int)(t & 15);
    float w = dis[i];
    agg1[t] = b1[ch] + t1[t] * w * w;
  }
}

// 16 lanes per edge, one channel each.
__global__ __launch_bounds__(256) void k_scatter1(const long long* __restrict__ ei,
                                                  const float* __restrict__ dis,
                                                  const float* __restrict__ t1,
                                                  float* __restrict__ agg1, long long E) {
  long long t = (long long)blockIdx.x * blockDim.x + threadIdx.x;
  long long e = t >> 4;
  int ch = (int)(t & 15);
  if (e < E) {
    long long s = ei[e];
    long long d = ei[E + e];
    float w = dis[s] * dis[d];
    atomicAdd(&agg1[d * HID + ch], t1[s * HID + ch] * w);
  }
}

// ---------- layer 2 GEMM: t2[N,32] = relu(agg1)[N,16] @ W2[16,32] ----------
// relu fused into the A-operand load. Two N-tiles -> two accumulators.
__global__ __launch_bounds__(256) void k_gemm2_wmma(const float* __restrict__ agg1,
                                                    const float* __restrict__ W2,
                                                    float* __restrict__ t2, int nTiles) {
  int wave = (int)((blockIdx.x * blockDim.x + threadIdx.x) >> 5);
  if (wave >= nTiles) return;
  int lane = threadIdx.x & 31;
  int m  = lane & 15;
  int hi = lane >> 4;

  v8f c0 = {}, c1 = {};
  const float* __restrict__ arow = agg1 + (size_t)(wave * 16 + m) * HID;
  #pragma unroll
  for (int kk = 0; kk < HID / 4; ++kk) {
    int k0 = kk * 4 + hi * 2;
    v2f a, bA, bB;
    a[0] = fmaxf(arow[k0], 0.0f);      // relu on h
    a[1] = fmaxf(arow[k0 + 1], 0.0f);
    bA[0] = W2[(size_t)k0 * OUT_CH + m];
    bA[1] = W2[(size_t)(k0 + 1) * OUT_CH + m];
    bB[0] = W2[(size_t)k0 * OUT_CH + 16 + m];
    bB[1] = W2[(size_t)(k0 + 1) * OUT_CH + 16 + m];
    c0 = __builtin_amdgcn_wmma_f32_16x16x4_f32(false, a, false, bA, (short)0, c0, false, false);
    c1 = __builtin_amdgcn_wmma_f32_16x16x4_f32(false, a, false, bB, (short)0, c1, false, false);
  }
  float* __restrict__ orow = t2 + (size_t)(wave * 16 + hi * 8) * OUT_CH;
  #pragma unroll
  for (int r = 0; r < 8; ++r) {
    orow[(size_t)r * OUT_CH + m]      = c0[r];
    orow[(size_t)r * OUT_CH + 16 + m] = c1[r];
  }
}

// ---------- layer 2 aggregation (into d_out) ----------
__global__ __launch_bounds__(256) void k_self2(const float* __restrict__ t2,
                                               const float* __restrict__ dis,
                                               const float* __restrict__ b2,
                                               float* __restrict__ out, long long total) {
  long long t = (long long)blockIdx.x * blockDim.x + threadIdx.x;
  if (t < total) {
    long long i = t >> 5;
    int ch = (int)(t & 31);
    float w = dis[i];
    out[t] = b2[ch] + t2[t] * w * w;
  }
}

// 32 lanes per edge (full wave), one channel each.
__global__ __launch_bounds__(256) void k_scatter2(const long long* __restrict__ ei,
                                                  const float* __restrict__ dis,
                                                  const float* __restrict__ t2,
                                                  float* __restrict__ out, long long E) {
  long long t = (long long)blockIdx.x * blockDim.x + threadIdx.x;
  long long e = t >> 5;
  int ch = (int)(t & 31);
  if (e < E) {
    long long s = ei[e];
    long long d = ei[E + e];
    float w = dis[s] * dis[d];
    atomicAdd(&out[d * OUT_CH + ch], t2[s * OUT_CH + ch] * w);
  }
}

// ---------- launch ----------

extern "C" void kernel_launch(void* const* d_in, const int* in_sizes, int n_in,
                              void* d_out, int out_size, void* d_ws, size_t ws_size,
                              hipStream_t stream) {
  const float*     x  = (const float*)d_in[0];
  const long long* ei = (const long long*)d_in[1];  // int64 [2, E]
  const float*     W1 = (const float*)d_in[2];
  const float*     b1 = (const float*)d_in[3];
  const float*     W2 = (const float*)d_in[4];
  const float*     b2 = (const float*)d_in[5];
  float* out = (float*)d_out;

  const int       N = in_sizes[0] / IN_CH;          // 100000
  const long long E = (long long)in_sizes[1] / 2;   // 6400000

  // workspace layout (floats): deg/dis [Npad] | t1 [N*16] | agg1 [N*16] | t2 [N*32]
  float* deg  = (float*)d_ws;
  size_t Npad = ((size_t)N + 63) & ~(size_t)63;
  float* t1   = deg + Npad;
  float* agg1 = t1 + (size_t)N * HID;
  float* t2   = agg1 + (size_t)N * HID;

  const int B = 256;
  const int nTiles = N / 16;  // N is an exact multiple of 16 for this problem

  k_init_deg <<<(N + B - 1) / B, B, 0, stream>>>(deg, N);
  k_count_deg<<<(unsigned)((E + B - 1) / B), B, 0, stream>>>(ei, deg, E);
  k_rsqrt_deg<<<(N + B - 1) / B, B, 0, stream>>>(deg, N);

  k_gemm1_wmma<<<(nTiles * 32 + B - 1) / B, B, 0, stream>>>(x, W1, t1, nTiles);
  k_self1   <<<(unsigned)(((long long)N * HID + B - 1) / B), B, 0, stream>>>(t1, deg, b1, agg1, (long long)N * HID);
  k_scatter1<<<(unsigned)((E * HID + B - 1) / B), B, 0, stream>>>(ei, deg, t1, agg1, E);

  k_gemm2_wmma<<<(nTiles * 32 + B - 1) / B, B, 0, stream>>>(agg1, W2, t2, nTiles);
  k_self2   <<<(unsigned)(((long long)N * OUT_CH + B - 1) / B), B, 0, stream>>>(t2, deg, b2, out, (long long)N * OUT_CH);
  k_scatter2<<<(unsigned)((E * OUT_CH + B - 1) / B), B, 0, stream>>>(ei, deg, t2, out, E);
}